// EdgeNetwork_61057255080035
// MI455X (gfx1250) — compile-verified
//
#include <hip/hip_runtime.h>
#include <hip/hip_bf16.h>

typedef __attribute__((ext_vector_type(2))) float v2f;
typedef __attribute__((ext_vector_type(8))) float v8f;

#define NATOMS 10000
#define NH     75          // hidden dim
#define NPF    14          // pair features
#define NP     15          // 14 pair features + 1 bias row
#define NZ     (NP * NH)   // 1125 columns of Z
#define NEDGES 64000

#define TILES_M (NATOMS / 16)        // 625
#define TILES_N ((NZ + 15) / 16)     // 71
#define KSTEPS  ((NH + 3) / 4)       // 19 (K padded 75 -> 76)

// ---------------------------------------------------------------------------
// Zero the output accumulator (segment_sum target).
// ---------------------------------------------------------------------------
__global__ void zero_out_kernel(float* __restrict__ out, int n) {
    int i = blockIdx.x * blockDim.x + threadIdx.x;
    if (i < n) out[i] = 0.0f;
}

// ---------------------------------------------------------------------------
// Prep: swizzle atom_features into WMMA A-fragment order.
// AFrag[((tm*KSTEPS + kk)*32 + lane)] = {AF[row, k], AF[row, k+1]}
//   row = tm*16 + (lane&15), k = kk*4 + (lane>>4)*2, zero-padded at k>=75.
// ---------------------------------------------------------------------------
__global__ void build_afrag_kernel(const float* __restrict__ AF, v2f* __restrict__ AFrag) {
    int idx = blockIdx.x * blockDim.x + threadIdx.x;
    if (idx >= TILES_M * KSTEPS * 32) return;
    const int lane = idx & 31;
    const int kk   = (idx >> 5) % KSTEPS;
    const int tm   = (idx >> 5) / KSTEPS;
    const int row  = tm * 16 + (lane & 15);
    const int k    = kk * 4 + (lane >> 4) * 2;
    v2f v;
    v.x = (k     < NH) ? AF[(size_t)row * NH + k]     : 0.f;
    v.y = (k + 1 < NH) ? AF[(size_t)row * NH + k + 1] : 0.f;
    AFrag[idx] = v;
}

// ---------------------------------------------------------------------------
// Prep: swizzle W2 into WMMA B-fragment order.
//   W2[k, col] with col = p*75+h:  (p<14) ? W[p*5625 + h*75 + k] : b[h*75+k]
// BFrag[((tn*KSTEPS + kk)*32 + lane)] = {W2[k, col], W2[k+1, col]}
//   col = tn*16 + (lane&15), k = kk*4 + (lane>>4)*2; zero-padded at
//   col>=1125 and k>=75.
// ---------------------------------------------------------------------------
__global__ void build_bfrag_kernel(const float* __restrict__ W,
                                   const float* __restrict__ bias,
                                   v2f* __restrict__ BFrag) {
    int idx = blockIdx.x * blockDim.x + threadIdx.x;
    if (idx >= TILES_N * KSTEPS * 32) return;
    const int lane = idx & 31;
    const int kk   = (idx >> 5) % KSTEPS;
    const int tn   = (idx >> 5) / KSTEPS;
    const int col  = tn * 16 + (lane & 15);
    const int k    = kk * 4 + (lane >> 4) * 2;
    v2f v = {0.f, 0.f};
    if (col < NZ) {
        const int p = col / NH;
        const int h = col % NH;
        const float* src = (p < NPF) ? (W + (size_t)p * (NH * NH) + (size_t)h * NH)
                                     : (bias + (size_t)h * NH);
        if (k     < NH) v.x = src[k];
        if (k + 1 < NH) v.y = src[k + 1];
    }
    BFrag[idx] = v;
}

// ---------------------------------------------------------------------------
// Phase 1: Z = AF @ W2 via f32 WMMA on pre-swizzled fragments.
// One wave per 16x16 output tile; inner loop = 2 coalesced b64 loads + 1 WMMA.
// ---------------------------------------------------------------------------
__global__ __launch_bounds__(256) void
phase1_gemm_kernel(const v2f* __restrict__ AFrag,
                   const v2f* __restrict__ BFrag,
                   float* __restrict__ Z)           // (10000, 1125)
{
    const int waveId = blockIdx.x * (256 / 32) + (threadIdx.x >> 5);
    if (waveId >= TILES_M * TILES_N) return;        // whole-wave uniform exit

    const int tm   = waveId / TILES_N;
    const int tn   = waveId % TILES_N;
    const int lane = threadIdx.x & 31;

    const v2f* ap = AFrag + (size_t)tm * KSTEPS * 32 + lane;
    const v2f* bp = BFrag + (size_t)tn * KSTEPS * 32 + lane;

    v8f acc = {0.f, 0.f, 0.f, 0.f, 0.f, 0.f, 0.f, 0.f};
#pragma unroll
    for (int kk = 0; kk < KSTEPS; ++kk) {
        // 8 args: (neg_a, A, neg_b, B, c_mod, C, reuse_a, reuse_b)
        acc = __builtin_amdgcn_wmma_f32_16x16x4_f32(
            false, ap[kk * 32], false, bp[kk * 32], (short)0, acc, false, false);
    }

    // D layout: VGPR r -> row tm*16 + r + 8*(lane>>4), col tn*16 + (lane&15)
    const int laneLo = lane & 15;
    const int laneHi = lane >> 4;
    const int colB   = tn * 16 + laneLo;
    if (colB < NZ) {
        float* zp = Z + (size_t)(tm * 16 + laneHi * 8) * NZ + colB;
#pragma unroll
        for (int r = 0; r < 8; ++r)
            zp[(size_t)r * NZ] = acc[r];
    }
}

// ---------------------------------------------------------------------------
// Phase 2: per edge e:  out[i_e, h] += sum_p pf'[e,p] * Z[j_e, p*75+h]
// One wave per edge; lane covers h = lane, lane+32, lane+64 (masked at 75).
// Segment sum via global f32 atomics (L2-resident target, 3 MB).
// ---------------------------------------------------------------------------
__global__ __launch_bounds__(256) void
phase2_edge_kernel(const float* __restrict__ PF,   // (64000, 14)
                   const int*   __restrict__ A2P,  // (64000, 2) int32
                   const float* __restrict__ Z,    // (10000, 1125)
                   float* __restrict__ out)        // (10000, 75)
{
    const int e = (blockIdx.x * blockDim.x + threadIdx.x) >> 5;
    if (e >= NEDGES) return;
    const int lane = threadIdx.x & 31;

    const int i = A2P[2 * e];       // segment target
    const int j = A2P[2 * e + 1];   // gathered atom

    const float* zrow = Z + (size_t)j * NZ;
    const int h0 = lane, h1 = lane + 32, h2 = lane + 64;

    float a0 = 0.f, a1 = 0.f, a2 = 0.f;
#pragma unroll
    for (int p = 0; p < NP; ++p) {
        const float w = (p < NPF) ? PF[(size_t)e * NPF + p] : 1.0f; // bias row
        const float* zp = zrow + p * NH;
        a0 += w * zp[h0];
        a1 += w * zp[h1];
        if (h2 < NH) a2 += w * zp[h2];
    }

    float* op = out + (size_t)i * NH;
    atomicAdd(op + h0, a0);
    atomicAdd(op + h1, a1);
    if (h2 < NH) atomicAdd(op + h2, a2);
}

// ---------------------------------------------------------------------------
extern "C" void kernel_launch(void* const* d_in, const int* in_sizes, int n_in,
                              void* d_out, int out_size, void* d_ws, size_t ws_size,
                              hipStream_t stream) {
    const float* pair_features = (const float*)d_in[0]; // (64000, 14)
    const float* atom_features = (const float*)d_in[1]; // (10000, 75)
    const int*   atom_to_pair  = (const int*)  d_in[2]; // (64000, 2)
    const float* W             = (const float*)d_in[3]; // (14, 5625)
    const float* b             = (const float*)d_in[4]; // (5625,)
    float* out = (float*)d_out;                         // (10000, 75)

    // Workspace layout (all offsets 8-byte aligned):
    //   Z     : 10000*1125 f32 = 45,000,000 B
    //   AFrag : 625*19*32 v2f  =  3,040,000 B
    //   BFrag :  71*19*32 v2f  =    345,344 B
    char* ws = (char*)d_ws;
    float* Z     = (float*)(ws);
    v2f*   AFrag = (v2f*)(ws + (size_t)NATOMS * NZ * sizeof(float));
    v2f*   BFrag = (v2f*)(ws + (size_t)NATOMS * NZ * sizeof(float)
                             + (size_t)TILES_M * KSTEPS * 32 * sizeof(v2f));

    // Zero segment-sum accumulator
    {
        const int n = NATOMS * NH;
        zero_out_kernel<<<(n + 255) / 256, 256, 0, stream>>>(out, n);
    }
    // Build WMMA fragments
    {
        const int nA = TILES_M * KSTEPS * 32;
        build_afrag_kernel<<<(nA + 255) / 256, 256, 0, stream>>>(atom_features, AFrag);
        const int nB = TILES_N * KSTEPS * 32;
        build_bfrag_kernel<<<(nB + 255) / 256, 256, 0, stream>>>(W, b, BFrag);
    }
    // Phase 1: per-atom GEMM via f32 WMMA
    {
        const int tiles = TILES_M * TILES_N;                  // 44375 waves
        const int wavesPerBlock = 256 / 32;
        const int blocks = (tiles + wavesPerBlock - 1) / wavesPerBlock;
        phase1_gemm_kernel<<<blocks, 256, 0, stream>>>(AFrag, BFrag, Z);
    }
    // Phase 2: edge gather + combine + atomic segment sum
    {
        const int wavesPerBlock = 256 / 32;
        const int blocks = (NEDGES + wavesPerBlock - 1) / wavesPerBlock;
        phase2_edge_kernel<<<blocks, 256, 0, stream>>>(pair_features, atom_to_pair, Z, out);
    }
}